// ManeuverHead_670014898684
// MI455X (gfx1250) — compile-verified
//
#include <hip/hip_runtime.h>

// ---------------------------------------------------------------------------
// ManeuverHead for MI455X (gfx1250).
// Only the first G=8 selected nodes per batch matter (<=32768 MLP rows).
// Exact segmented prefix scan assigns ranks; a 16-row-per-wave fp32 WMMA
// MLP (V_WMMA_F32_16X16X4_F32) computes both layers. Tile staging uses
// CDNA5 GLOBAL_LOAD_ASYNC_TO_LDS_B128 (ASYNCcnt) when available to avoid
// the global->VGPR->LDS round trip and keep VGPR pressure low.
// ---------------------------------------------------------------------------

#define N_NODES   262144
#define B_BATCH   4096
#define G_GRP     8
#define D_DIM     7
#define NODE_DIM  128
#define GLOB_DIM  128
#define HIDDEN    64
#define NEG_F     (-1000000000.0f)
#define NSLOTS    (B_BATCH * G_GRP)          // 32768
#define OUT_ELEMS (B_BATCH * G_GRP * D_DIM)  // 229376

typedef float v2f __attribute__((ext_vector_type(2)));
typedef float v4f __attribute__((ext_vector_type(4)));
typedef float v8f __attribute__((ext_vector_type(8)));
typedef int   v4i __attribute__((ext_vector_type(4)));

#if defined(__has_builtin)
#if __has_builtin(__builtin_amdgcn_global_load_async_to_lds_b128)
#define HAVE_ASYNC_LDS 1
#endif
#endif

#ifdef HAVE_ASYNC_LDS
typedef __attribute__((address_space(1))) v4i gas_v4i;  // global (AS1)
typedef __attribute__((address_space(3))) v4i las_v4i;  // LDS (AS3)
__device__ __forceinline__ void async_ld_b128(const float* g, float* l) {
  // mem -> LDS, 16B per lane, no data VGPRs; tracked by ASYNCcnt.
  // AS3 pointer = low 32 bits of the flat LDS address (ISA aperture rule).
  gas_v4i* gp = (gas_v4i*)(uintptr_t)g;
  las_v4i* lp = (las_v4i*)(uintptr_t)l;
  __builtin_amdgcn_global_load_async_to_lds_b128(gp, lp, 0, 0);
}
__device__ __forceinline__ void async_wait0() {
  asm volatile("s_wait_asynccnt 0x0" ::: "memory");
}
#endif

// ---------------- Kernel 1: block partial sums + init --------------------
__global__ void k_reduce_init(const unsigned char* __restrict__ sel,
                              int* __restrict__ partial,
                              int* __restrict__ slot_node,
                              float* __restrict__ out) {
  int t  = threadIdx.x;
  int gt = blockIdx.x * 256 + t;      // 0..65535
  int base = gt * 4;
  int s = 0;
#pragma unroll
  for (int j = 0; j < 4; ++j) s += sel[base + j] ? 1 : 0;
  __shared__ int red[256];
  red[t] = s;
  __syncthreads();
  for (int d = 128; d > 0; d >>= 1) {
    if (t < d) red[t] += red[t + d];
    __syncthreads();
  }
  if (t == 0) partial[blockIdx.x] = red[0];
  if (gt < NSLOTS) slot_node[gt] = -1;
#pragma unroll
  for (int j = 0; j < 4; ++j) {
    int oi = base + j;
    if (oi < OUT_ELEMS) out[oi] = NEG_F;
  }
}

// ---------------- Kernel 2: scan the 256 block partials ------------------
__global__ void k_scan256(int* __restrict__ partial) {
  __shared__ int s[256];
  int t = threadIdx.x;
  s[t] = partial[t];
  __syncthreads();
  for (int d = 1; d < 256; d <<= 1) {
    int v = (t >= d) ? s[t - d] : 0;
    __syncthreads();
    s[t] += v;
    __syncthreads();
  }
  partial[t] = (t == 0) ? 0 : s[t - 1];   // exclusive block offsets
}

// ---------------- Kernel 3: inclusive csum + batch-start offsets ---------
__global__ void k_scan_mark(const unsigned char* __restrict__ sel,
                            const int* __restrict__ batch,
                            const int* __restrict__ blockExcl,
                            int* __restrict__ csum,
                            int* __restrict__ offsets) {
  int t = threadIdx.x;
  int base = (blockIdx.x * 256 + t) * 4;
  int sv[4];
  int s = 0;
#pragma unroll
  for (int j = 0; j < 4; ++j) { sv[j] = sel[base + j] ? 1 : 0; s += sv[j]; }
  __shared__ int sc[256];
  sc[t] = s;
  __syncthreads();
  for (int d = 1; d < 256; d <<= 1) {
    int v = (t >= d) ? sc[t - d] : 0;
    __syncthreads();
    sc[t] += v;
    __syncthreads();
  }
  int run = blockExcl[blockIdx.x] + (t ? sc[t - 1] : 0);
#pragma unroll
  for (int j = 0; j < 4; ++j) {
    int i = base + j;
    int excl = run;
    run += sv[j];
    csum[i] = run;
    int b = batch[i];
    if (i == 0 || batch[i - 1] != b) offsets[b] = excl;  // selected-before-batch
  }
}

// ---------------- Kernel 4: fill slot table ------------------------------
__global__ void k_slots(const unsigned char* __restrict__ sel,
                        const int* __restrict__ batch,
                        const int* __restrict__ csum,
                        const int* __restrict__ offsets,
                        int* __restrict__ slot_node) {
  int i = blockIdx.x * 256 + threadIdx.x;
  if (i >= N_NODES || !sel[i]) return;
  int b = batch[i];
  int g = csum[i] - 1 - offsets[b];
  if (g < G_GRP) slot_node[b * G_GRP + g] = i;
}

// ---------------- Kernel 5: fp32 WMMA MLP over 16-slot tiles -------------
// 2 waves / block, 1 tile (M=16 slots) per wave. 1024 blocks -> 2048 tiles.
// GEMM1: M16 N64 K256 via V_WMMA_F32_16X16X4_F32, K staged in two 128-halves.
// GEMM2: M16 N16(pad of 7) K64.
__global__ __launch_bounds__(64)
void k_mlp(const float* __restrict__ nodef, const float* __restrict__ globf,
           const unsigned char* __restrict__ mmask,
           const float* __restrict__ W1, const float* __restrict__ b1,
           const float* __restrict__ W2, const float* __restrict__ b2,
           const int* __restrict__ slot_node, float* __restrict__ out) {
  __shared__ float W1s[128 * HIDDEN];     // 32 KB: one K-half of W1 [k][n]
  __shared__ float Ats[2][16 * 128];      // 16 KB: per-wave A tile half [m][k]
  __shared__ float b1s[HIDDEN];
  __shared__ float b2s[16];
  float* W2s = W1s;                       // W2 (64x16, n>=7 zero) reuses W1s

  int tid  = threadIdx.x;
  int wave = tid >> 5;
  int lane = tid & 31;
  int tile = blockIdx.x * 2 + wave;
  float* At = Ats[wave];

  int m    = lane & 15;           // row of A fragment
  int khi  = (lane >> 4) << 1;    // lane-half K offset: 0 or 2
  int ncol = lane & 15;           // column of B/C/D fragment

  if (tid < HIDDEN) b1s[tid] = b1[tid];
  if (tid < 16)     b2s[tid] = (tid < D_DIM) ? b2[tid] : 0.0f;

  v8f acc0 = {}, acc1 = {}, acc2 = {}, acc3 = {};

#pragma unroll
  for (int pass = 0; pass < 2; ++pass) {
#ifdef HAVE_ASYNC_LDS
    // --- CDNA5 async staging: mem -> LDS with no data VGPRs -------------
    {
      const float* src = W1 + pass * 128 * HIDDEN;   // 8192 floats this half
#pragma unroll
      for (int i = 0; i < 32; ++i)
        async_ld_b128(src + (tid + i * 64) * 4, W1s + (tid + i * 64) * 4);
    }
    for (int r = 0; r < 16; ++r) {
      int slot = tile * 16 + r;
      int node = slot_node[slot];
      int bb   = slot >> 3;
      int nd   = node < 0 ? 0 : node;   // clamp: empty rows never written out
      const float* src = (pass == 0) ? (nodef + nd * NODE_DIM)
                                     : (globf + bb * GLOB_DIM);
      async_ld_b128(src + lane * 4, At + r * 128 + lane * 4);
    }
    async_wait0();       // own wave's ASYNCcnt -> 0 (LDS writes landed)
    __syncthreads();     // make all waves' staging visible block-wide
#else
    // --- fallback: VGPR round-trip staging ------------------------------
    {
      const v4f* src = (const v4f*)(W1 + pass * 128 * HIDDEN);
      v4f* dst = (v4f*)W1s;
#pragma unroll
      for (int i = 0; i < 32; ++i) dst[tid + i * 64] = src[tid + i * 64];
    }
    for (int r = 0; r < 16; ++r) {
      int slot = tile * 16 + r;
      int node = slot_node[slot];
      int bb   = slot >> 3;
      int nd   = node < 0 ? 0 : node;
      const v4f* src = (pass == 0) ? (const v4f*)(nodef + nd * NODE_DIM)
                                   : (const v4f*)(globf + bb * GLOB_DIM);
      ((v4f*)(At + r * 128))[lane] = src[lane];
    }
    __syncthreads();
#endif

    // 32 k-steps of 16x16x4 fp32 WMMA over 4 N-tiles
#pragma unroll 2
    for (int kt = 0; kt < 32; ++kt) {
      int k0 = kt * 4 + khi;
      v2f a;
      a.x = At[m * 128 + k0];
      a.y = At[m * 128 + k0 + 1];
      v2f bf;
      bf.x = W1s[k0 * HIDDEN + ncol];
      bf.y = W1s[(k0 + 1) * HIDDEN + ncol];
      acc0 = __builtin_amdgcn_wmma_f32_16x16x4_f32(false, a, false, bf, (short)0, acc0, false, false);
      bf.x = W1s[k0 * HIDDEN + 16 + ncol];
      bf.y = W1s[(k0 + 1) * HIDDEN + 16 + ncol];
      acc1 = __builtin_amdgcn_wmma_f32_16x16x4_f32(false, a, false, bf, (short)0, acc1, false, false);
      bf.x = W1s[k0 * HIDDEN + 32 + ncol];
      bf.y = W1s[(k0 + 1) * HIDDEN + 32 + ncol];
      acc2 = __builtin_amdgcn_wmma_f32_16x16x4_f32(false, a, false, bf, (short)0, acc2, false, false);
      bf.x = W1s[k0 * HIDDEN + 48 + ncol];
      bf.y = W1s[(k0 + 1) * HIDDEN + 48 + ncol];
      acc3 = __builtin_amdgcn_wmma_f32_16x16x4_f32(false, a, false, bf, (short)0, acc3, false, false);
    }
    __syncthreads();   // all waves done with this W1 half before reload
  }

  // stage W2 padded to 16 cols into the (now free) W1s region
  for (int idx = tid; idx < HIDDEN * 16; idx += 64) {
    int k = idx >> 4, n = idx & 15;
    W2s[idx] = (n < D_DIM) ? W2[k * D_DIM + n] : 0.0f;
  }
  // bias+relu, write H (16x64) back into own A-tile region (same-wave DS is
  // in-order; barrier below also covers W2s visibility)
  int mbase = (lane >> 4) * 8;   // C/D layout: lanes 16-31 hold rows M+8
#pragma unroll
  for (int r = 0; r < 8; ++r) {
    int mr = mbase + r;
    At[mr * HIDDEN + ncol]      = fmaxf(acc0[r] + b1s[ncol],      0.0f);
    At[mr * HIDDEN + 16 + ncol] = fmaxf(acc1[r] + b1s[16 + ncol], 0.0f);
    At[mr * HIDDEN + 32 + ncol] = fmaxf(acc2[r] + b1s[32 + ncol], 0.0f);
    At[mr * HIDDEN + 48 + ncol] = fmaxf(acc3[r] + b1s[48 + ncol], 0.0f);
  }
  __syncthreads();

  // GEMM2: out = H @ W2  (M16, N16(pad), K64)
  v8f dacc = {};
#pragma unroll
  for (int kt = 0; kt < 16; ++kt) {
    int k0 = kt * 4 + khi;
    v2f a;
    a.x = At[m * HIDDEN + k0];
    a.y = At[m * HIDDEN + k0 + 1];
    v2f bf;
    bf.x = W2s[k0 * 16 + ncol];
    bf.y = W2s[(k0 + 1) * 16 + ncol];
    dacc = __builtin_amdgcn_wmma_f32_16x16x4_f32(false, a, false, bf, (short)0, dacc, false, false);
  }

  // scatter: only occupied slots, only where maneuver_mask is set
  if (ncol < D_DIM) {
    float bias = b2s[ncol];
#pragma unroll
    for (int r = 0; r < 8; ++r) {
      int mr   = mbase + r;
      int slot = tile * 16 + mr;
      if (slot_node[slot] >= 0) {
        int oi = slot * D_DIM + ncol;
        if (mmask[oi]) out[oi] = dacc[r] + bias;
      }
    }
  }
}

// ---------------------------------------------------------------------------
extern "C" void kernel_launch(void* const* d_in, const int* in_sizes, int n_in,
                              void* d_out, int out_size, void* d_ws, size_t ws_size,
                              hipStream_t stream) {
  const float*         nodef = (const float*)d_in[0];
  const float*         globf = (const float*)d_in[1];
  const unsigned char* sel   = (const unsigned char*)d_in[2]; // jnp bool = 1 byte
  const int*           batch = (const int*)d_in[3];
  const unsigned char* mmask = (const unsigned char*)d_in[4]; // jnp bool = 1 byte
  const float*         W1    = (const float*)d_in[5];
  const float*         b1    = (const float*)d_in[6];
  const float*         W2    = (const float*)d_in[7];
  const float*         b2    = (const float*)d_in[8];
  float* out = (float*)d_out;

  int* ws        = (int*)d_ws;
  int* partial   = ws;                             // 256
  int* offsets   = ws + 256;                       // 4096
  int* slot_node = ws + 256 + B_BATCH;             // 32768
  int* csum      = ws + 256 + B_BATCH + NSLOTS;    // 262144   (~1.2 MB total)

  k_reduce_init<<<256, 256, 0, stream>>>(sel, partial, slot_node, out);
  k_scan256<<<1, 256, 0, stream>>>(partial);
  k_scan_mark<<<256, 256, 0, stream>>>(sel, batch, partial, csum, offsets);
  k_slots<<<N_NODES / 256, 256, 0, stream>>>(sel, batch, csum, offsets, slot_node);
  k_mlp<<<NSLOTS / 32, 64, 0, stream>>>(nodef, globf, mmask, W1, b1, W2, b2,
                                        slot_node, out);
}